// MambaBlock_13211319402874
// MI455X (gfx1250) — compile-verified
//
#include <hip/hip_runtime.h>
#include <hip/hip_bf16.h>
#include <math.h>

typedef __attribute__((ext_vector_type(2))) float v2f;
typedef __attribute__((ext_vector_type(8))) float v8f;
typedef __attribute__((ext_vector_type(2))) int   v2i;

#define D_MODEL 1024
#define D_STATE 16
#define D_INNER 2048
#define DT_RANK 64
#define BSZ 2
#define LSEQ 2048
#define NROWS (BSZ * LSEQ)   // 4096

// gfx1250 async global->LDS copies (ASYNCcnt-tracked). Guarded: fall back to
// synchronous copies if this toolchain doesn't declare the builtins.
#if defined(__AMDGCN__) && defined(__gfx1250__) && \
    __has_builtin(__builtin_amdgcn_global_load_async_to_lds_b64) && \
    __has_builtin(__builtin_amdgcn_global_load_async_to_lds_b32)
#define HAVE_ASYNC_LDS 1
#define GLOBAL_AS __attribute__((address_space(1)))
#define LDS_AS __attribute__((address_space(3)))
#else
#define HAVE_ASYNC_LDS 0
#endif

__device__ __forceinline__ v8f wmma_f32x4(v2f a, v2f b, v8f c) {
    // D = A(16x4,f32) * B(4x16,f32) + C(16x16,f32)
    return __builtin_amdgcn_wmma_f32_16x16x4_f32(false, a, false, b, (short)0, c,
                                                 false, false);
}

__device__ __forceinline__ float softplusf(float x) {
    return (x > 20.f) ? x : log1pf(__expf(x));
}

// ---------------------------------------------------------------------------
// 32x32 LDS transpose: out[C x R] = in[R x C]^T   (R, C multiples of 32)
// ---------------------------------------------------------------------------
__global__ void transpose_k(const float* __restrict__ in, float* __restrict__ out,
                            int R, int C) {
    __shared__ float t[32][33];
    int bx = blockIdx.x * 32, by = blockIdx.y * 32;
    for (int j = 0; j < 32; j += 8)
        t[threadIdx.y + j][threadIdx.x] =
            in[(size_t)(by + threadIdx.y + j) * C + (bx + threadIdx.x)];
    __syncthreads();
    for (int j = 0; j < 32; j += 8)
        out[(size_t)(bx + threadIdx.y + j) * R + (by + threadIdx.x)] =
            t[threadIdx.x][threadIdx.y + j];
}

// ---------------------------------------------------------------------------
// Tiled fp32 WMMA GEMM: D[M x N] = A[M x K] * Bt[K x N]
// Block = 256 threads = 8 waves. Block tile 128x64, wave tile 16x64.
// Double-buffered LDS staging; async global->LDS copies (ASYNCcnt).
// sA row-major stride 18, sB column-major stride 18: every WMMA fragment is
// one aligned, bank-conflict-free ds_load_b64.
// mode 0: plain store; mode 1: D = softplus(acc + bias[n])  (dt projection)
// ---------------------------------------------------------------------------
#define KC 16
#define SA_STRIDE 18                 // 128 rows x 18
#define SB_STRIDE 18                 // 64 cols  x 18

__device__ __forceinline__ void stage_chunk(const float* __restrict__ A,
                                            const float* __restrict__ Bt,
                                            float* sAb, float* sBb,
                                            int m0, int n0, int k0,
                                            int lda, int ldb, int tid) {
#if HAVE_ASYNC_LDS
    // A: 128x16 fp32 as 64-bit async copies (both sides 8B aligned)
    for (int i = tid; i < 128 * 8; i += 256) {
        int r = i >> 3, c2 = (i & 7) * 2;
        __builtin_amdgcn_global_load_async_to_lds_b64(
            (GLOBAL_AS v2i*)(A + (size_t)(m0 + r) * lda + k0 + c2),
            (LDS_AS v2i*)(sAb + r * SA_STRIDE + c2), 0, 0);
    }
    // B: 16x64 fp32, scattered into column-major LDS -> 32-bit async copies
    for (int i = tid; i < 16 * 64; i += 256) {
        int r = i >> 6, c = i & 63;
        __builtin_amdgcn_global_load_async_to_lds_b32(
            (GLOBAL_AS int*)(Bt + (size_t)(k0 + r) * ldb + n0 + c),
            (LDS_AS int*)(sBb + c * SB_STRIDE + r), 0, 0);
    }
#else
    for (int i = tid; i < 128 * 16; i += 256) {
        int r = i >> 4, c = i & 15;
        sAb[r * SA_STRIDE + c] = A[(size_t)(m0 + r) * lda + (k0 + c)];
    }
    for (int i = tid; i < 16 * 64; i += 256) {
        int r = i >> 6, c = i & 63;
        sBb[c * SB_STRIDE + r] = Bt[(size_t)(k0 + r) * ldb + (n0 + c)];
    }
#endif
}

__device__ __forceinline__ void stage_join() {
#if HAVE_ASYNC_LDS
    asm volatile("s_wait_asynccnt 0x0" ::: "memory");
#endif
    __syncthreads();
}

__global__ void gemm_wmma_tiled(const float* __restrict__ A,
                                const float* __restrict__ Bt,
                                float* __restrict__ Dst,
                                const float* __restrict__ bias,
                                int K, int lda, int ldb, int ldd, int mode) {
    __shared__ float sA[2][128 * SA_STRIDE];
    __shared__ float sB[2][64 * SB_STRIDE];
    const int tid  = threadIdx.x;
    const int lane = tid & 31;
    const int wave = tid >> 5;
    const int m0 = blockIdx.y * 128;
    const int n0 = blockIdx.x * 64;
    const int row = lane & 15;       // M (A) / N (B,C) index
    const int hi  = lane >> 4;       // half-wave selector

    v8f acc[4];
    v8f zero = {};
    acc[0] = zero; acc[1] = zero; acc[2] = zero; acc[3] = zero;

    const int nchunks = K / KC;
    int p = 0;
    stage_chunk(A, Bt, sA[0], sB[0], m0, n0, 0, lda, ldb, tid);
    stage_join();

    for (int c = 0; c < nchunks; ++c) {
        if (c + 1 < nchunks)
            stage_chunk(A, Bt, sA[p ^ 1], sB[p ^ 1], m0, n0, (c + 1) * KC,
                        lda, ldb, tid);
        const float* sAb = sA[p];
        const float* sBb = sB[p];
#pragma unroll
        for (int kk = 0; kk < KC; kk += 4) {
            v2f a = *reinterpret_cast<const v2f*>(
                sAb + (wave * 16 + row) * SA_STRIDE + kk + 2 * hi);
#pragma unroll
            for (int j = 0; j < 4; ++j) {
                v2f b = *reinterpret_cast<const v2f*>(
                    sBb + (j * 16 + row) * SB_STRIDE + kk + 2 * hi);
                acc[j] = wmma_f32x4(a, b, acc[j]);
            }
        }
        stage_join();
        p ^= 1;
    }

#pragma unroll
    for (int j = 0; j < 4; ++j) {
        const int n = n0 + j * 16 + row;
        const float bv = (mode == 1) ? bias[n] : 0.f;
#pragma unroll
        for (int r = 0; r < 8; ++r) {
            float v = acc[j][r];
            if (mode == 1) v = softplusf(v + bv);
            Dst[(size_t)(m0 + wave * 16 + r + 8 * hi) * ldd + n] = v;
        }
    }
}

// ---------------------------------------------------------------------------
// Direct (register-only) 16x16-per-wave WMMA GEMM for skinny N (x_proj, N=96).
// ---------------------------------------------------------------------------
__global__ void gemm_wmma_16(const float* __restrict__ A,
                             const float* __restrict__ Bt,
                             float* __restrict__ Dst,
                             int K, int lda, int ldb, int ldd) {
    const int lane = threadIdx.x & 31;
    const int wave = threadIdx.x >> 5;
    const int mt = blockIdx.y * 8 + wave;
    const int nt = blockIdx.x;
    const int row = lane & 15;
    const int hi  = lane >> 4;
    const float* Ap = A + (size_t)(mt * 16 + row) * lda;
    const float* Bp = Bt + nt * 16 + row;
    v8f acc = {};
    for (int k = 0; k < K; k += 4) {
        v2f a, b;
        a.x = Ap[k + 2 * hi];
        a.y = Ap[k + 2 * hi + 1];
        b.x = Bp[(size_t)(k + 2 * hi) * ldb];
        b.y = Bp[(size_t)(k + 2 * hi + 1) * ldb];
        acc = wmma_f32x4(a, b, acc);
    }
#pragma unroll
    for (int r = 0; r < 8; ++r)
        Dst[(size_t)(mt * 16 + r + 8 * hi) * ldd + nt * 16 + row] = acc[r];
}

// ---------------------------------------------------------------------------
// Causal depthwise conv (D_CONV=4) + bias + SiLU.  xi lives in xz[:, :, 0:2048].
// ---------------------------------------------------------------------------
__global__ void conv_silu_k(const float* __restrict__ xz,
                            const float* __restrict__ cw,
                            const float* __restrict__ cb,
                            float* __restrict__ xc) {
    size_t idx = (size_t)blockIdx.x * 256 + threadIdx.x;  // over NROWS*D_INNER
    int d = (int)(idx & (D_INNER - 1));
    size_t bl = idx >> 11;            // b*LSEQ + l
    int l = (int)(bl & (LSEQ - 1));
    float acc = cb[d];
#pragma unroll
    for (int j = 0; j < 4; ++j) {
        int ls = l - 3 + j;
        if (ls >= 0)
            acc += xz[(bl - 3 + j) * (size_t)(2 * D_INNER) + d] * cw[d * 4 + j];
    }
    float sig = 1.f / (1.f + __expf(-acc));
    xc[idx] = acc * sig;
}

// ---------------------------------------------------------------------------
// Selective scan: one lane per (b, d, s).  y reduced over s via shfl_xor(16).
// Fuses the D*u skip connection and the silu(z) gate.
// ---------------------------------------------------------------------------
__global__ void mamba_scan_k(const float* __restrict__ dtb,
                             const float* __restrict__ xc,
                             const float* __restrict__ xdbl,
                             const float* __restrict__ xz,
                             const float* __restrict__ A_log,
                             const float* __restrict__ Dp,
                             float* __restrict__ yb) {
    const int s  = threadIdx.x & 15;
    const int dl = threadIdx.x >> 4;                 // 0..15
    const int d  = blockIdx.x * 16 + dl;             // 0..2047
    const int b  = blockIdx.y;
    const float Aneg = -__expf(A_log[d * D_STATE + s]);
    const float Dd = Dp[d];
    float h = 0.f;
    for (int l = 0; l < LSEQ; ++l) {
        const size_t rowi = (size_t)(b * LSEQ + l);
        float dt = dtb[rowi * D_INNER + d];          // broadcast across 16 lanes
        float u  = xc [rowi * D_INNER + d];
        float Bs = xdbl[rowi * 96 + 64 + s];
        float Cs = xdbl[rowi * 96 + 80 + s];
        float dA = __expf(dt * Aneg);
        h = dA * h + (dt * u) * Bs;
        float p = h * Cs;
#pragma unroll
        for (int o = 8; o > 0; o >>= 1) p += __shfl_xor(p, o, 16);
        if (s == 0) {
            float yv = p + Dd * u;
            float z = xz[rowi * (size_t)(2 * D_INNER) + D_INNER + d];
            float zs = z / (1.f + __expf(-z));
            yb[rowi * D_INNER + d] = yv * zs;
        }
    }
}

// ---------------------------------------------------------------------------
// In-place residual add + LayerNorm over the last dim (1024), 1 block per row.
// ---------------------------------------------------------------------------
__global__ void add_layernorm_k(float* __restrict__ out,
                                const float* __restrict__ x,
                                const float* __restrict__ g,
                                const float* __restrict__ beta) {
    const size_t base = (size_t)blockIdx.x * D_MODEL;
    float v[4], s = 0.f, sq = 0.f;
#pragma unroll
    for (int i = 0; i < 4; ++i) {
        int c = threadIdx.x + i * 256;
        float t = out[base + c] + x[base + c];
        v[i] = t; s += t; sq += t * t;
    }
#pragma unroll
    for (int o = 16; o > 0; o >>= 1) {
        s  += __shfl_xor(s,  o, 32);
        sq += __shfl_xor(sq, o, 32);
    }
    __shared__ float rs[8], rq[8];
    const int wave = threadIdx.x >> 5, lane = threadIdx.x & 31;
    if (lane == 0) { rs[wave] = s; rq[wave] = sq; }
    __syncthreads();
    if (threadIdx.x == 0) {
        float ts = 0.f, tq = 0.f;
        for (int w = 0; w < 8; ++w) { ts += rs[w]; tq += rq[w]; }
        rs[0] = ts; rq[0] = tq;
    }
    __syncthreads();
    const float mean = rs[0] * (1.f / D_MODEL);
    const float var  = rq[0] * (1.f / D_MODEL) - mean * mean;
    const float rstd = rsqrtf(var + 1e-5f);
#pragma unroll
    for (int i = 0; i < 4; ++i) {
        int c = threadIdx.x + i * 256;
        out[base + c] = (v[i] - mean) * rstd * g[c] + beta[c];
    }
}

// ---------------------------------------------------------------------------
extern "C" void kernel_launch(void* const* d_in, const int* in_sizes, int n_in,
                              void* d_out, int out_size, void* d_ws, size_t ws_size,
                              hipStream_t stream) {
    const float* x         = (const float*)d_in[0];
    const float* in_proj_w = (const float*)d_in[1];
    const float* conv_w    = (const float*)d_in[2];
    const float* conv_b    = (const float*)d_in[3];
    const float* x_proj_w  = (const float*)d_in[4];
    const float* dt_proj_w = (const float*)d_in[5];
    const float* dt_proj_b = (const float*)d_in[6];
    const float* A_log     = (const float*)d_in[7];
    const float* Dp        = (const float*)d_in[8];
    const float* out_proj_w= (const float*)d_in[9];
    const float* ln_g      = (const float*)d_in[10];
    const float* ln_b      = (const float*)d_in[11];
    float* out = (float*)d_out;

    float* ws = (float*)d_ws;
    size_t off = 0;
    float* wt_in  = ws + off; off += (size_t)D_MODEL * (2 * D_INNER);      // 1024 x 4096
    float* wt_xp  = ws + off; off += (size_t)D_INNER * 96;                 // 2048 x 96
    float* wt_dt  = ws + off; off += (size_t)DT_RANK * D_INNER;            // 64 x 2048
    float* wt_out = ws + off; off += (size_t)D_INNER * D_MODEL;            // 2048 x 1024
    float* xz     = ws + off; off += (size_t)NROWS * (2 * D_INNER);        // 4096 x 4096
    float* xc     = ws + off; off += (size_t)NROWS * D_INNER;              // 4096 x 2048
    float* xdbl   = ws + off; off += (size_t)NROWS * 96;                   // 4096 x 96
    float* dtb    = ws + off; off += (size_t)NROWS * D_INNER;              // 4096 x 2048
    float* yb     = ws + off; off += (size_t)NROWS * D_INNER;              // 4096 x 2048

    const dim3 tb(32, 8);
    // Weight transposes (so WMMA B-fragment loads are lane-coalesced)
    transpose_k<<<dim3(D_MODEL / 32, (2 * D_INNER) / 32), tb, 0, stream>>>(
        in_proj_w, wt_in, 2 * D_INNER, D_MODEL);
    transpose_k<<<dim3(D_INNER / 32, 96 / 32), tb, 0, stream>>>(
        x_proj_w, wt_xp, 96, D_INNER);
    transpose_k<<<dim3(DT_RANK / 32, D_INNER / 32), tb, 0, stream>>>(
        dt_proj_w, wt_dt, D_INNER, DT_RANK);
    transpose_k<<<dim3(D_INNER / 32, D_MODEL / 32), tb, 0, stream>>>(
        out_proj_w, wt_out, D_MODEL, D_INNER);

    // 1) in_proj: xz[4096 x 4096] = x[4096 x 1024] * wt_in[1024 x 4096]
    gemm_wmma_tiled<<<dim3((2 * D_INNER) / 64, NROWS / 128), 256, 0, stream>>>(
        x, wt_in, xz, nullptr, D_MODEL, D_MODEL, 2 * D_INNER, 2 * D_INNER, 0);

    // 2) causal depthwise conv + SiLU -> xc
    conv_silu_k<<<(NROWS * D_INNER) / 256, 256, 0, stream>>>(xz, conv_w, conv_b, xc);

    // 3) x_proj: xdbl[4096 x 96] = xc[4096 x 2048] * wt_xp[2048 x 96]
    gemm_wmma_16<<<dim3(96 / 16, NROWS / 128), 256, 0, stream>>>(
        xc, wt_xp, xdbl, D_INNER, D_INNER, 96, 96);

    // 4) dt = softplus(xdbl[:, :64] * wt_dt + dt_proj_b)  (lda = 96!)
    gemm_wmma_tiled<<<dim3(D_INNER / 64, NROWS / 128), 256, 0, stream>>>(
        xdbl, wt_dt, dtb, dt_proj_b, DT_RANK, 96, D_INNER, D_INNER, 1);

    // 5) selective scan + D*u skip + silu(z) gate -> yb
    mamba_scan_k<<<dim3(D_INNER / 16, BSZ), 256, 0, stream>>>(
        dtb, xc, xdbl, xz, A_log, Dp, yb);

    // 6) out_proj: out[4096 x 1024] = yb[4096 x 2048] * wt_out[2048 x 1024]
    gemm_wmma_tiled<<<dim3(D_MODEL / 64, NROWS / 128), 256, 0, stream>>>(
        yb, wt_out, out, nullptr, D_INNER, D_INNER, D_MODEL, D_MODEL, 0);

    // 7) residual + LayerNorm, in place on d_out
    add_layernorm_k<<<NROWS, 256, 0, stream>>>(out, x, ln_g, ln_b);
}